// HSTU_BSA_Triton_87170656240258
// MI455X (gfx1250) — compile-verified
//
#include <hip/hip_runtime.h>
#include <math.h>

#define BB   4
#define NN   1024
#define HH   8
#define DD   64
#define BS   32          // BLOCK_SIZE
#define NB   32          // N / BS
#define TOPS 16          // top-k S
#define QT   16          // query tile (WMMA M)
#define NQT  (NN / QT)   // 64 tiles per (b,h)

#define NEG_INF (-3.0e38f)

typedef __attribute__((ext_vector_type(2))) float v2f;
typedef __attribute__((ext_vector_type(8))) float v8f;

// D(16x16,f32) = A(16x4,f32) * B(4x16,f32) + C   — exact fp32 matrix op
__device__ __forceinline__ v8f wmma4(v2f a, v2f b, v8f c) {
  return __builtin_amdgcn_wmma_f32_16x16x4_f32(
      /*neg_a=*/false, a, /*neg_b=*/false, b,
      /*c_mod=*/(short)0, c, /*reuse_a=*/false, /*reuse_b=*/false);
}

__device__ __forceinline__ float silu_f(float x) {
  return x / (1.0f + __expf(-x));
}

// gfx1250 async DMA: global -> LDS, 16B per lane, tracked by ASYNCcnt.
// lds_addr is the 32-bit LDS byte offset (low 32 bits of the flat shared addr).
__device__ __forceinline__ void async_ld16(unsigned lds_addr, const void* gaddr) {
  asm volatile("global_load_async_to_lds_b128 %0, %1, off"
               :: "v"(lds_addr), "v"(gaddr) : "memory");
}
__device__ __forceinline__ void wait_async0() {
  asm volatile("s_wait_asynccnt 0x0" ::: "memory");
}
__device__ __forceinline__ unsigned lds_off(const void* p) {
  return (unsigned)(unsigned long long)p;   // LDS aperture: low 32 bits = offset
}

// ---------------- Kernel 1: block-mean K/V  ->  [B][H][nb][D] ----------------
__global__ void kcmp_vcmp_kernel(const float* __restrict__ pk,
                                 const float* __restrict__ pv,
                                 float* __restrict__ kc,
                                 float* __restrict__ vc) {
  int t = blockIdx.x * blockDim.x + threadIdx.x;   // t == output index
  if (t >= BB * HH * NB * DD) return;
  int d  = t & (DD - 1);
  int jb = (t >> 6) & (NB - 1);
  int h  = (t >> 11) & (HH - 1);
  int b  = t >> 14;
  float sk = 0.0f, sv = 0.0f;
  int base = ((b * NN + jb * BS) * HH + h) * DD + d;
  for (int s = 0; s < BS; ++s) {
    sk += pk[base + s * HH * DD];
    sv += pv[base + s * HH * DD];
  }
  kc[t] = sk * (1.0f / BS);
  vc[t] = sv * (1.0f / BS);
}

// ---------------- Kernel 2: sigmoid gates (g_cmp, g_slc) --------------------
__global__ void gates_kernel(const float* __restrict__ pq,
                             const float* __restrict__ gw,
                             float* __restrict__ gcmp,
                             float* __restrict__ gslc) {
  int t = blockIdx.x * blockDim.x + threadIdx.x;   // (b*N + n)*H + h
  if (t >= BB * NN * HH) return;
  int h = t % HH;
  const float* q = pq + (size_t)t * DD;
  float s0 = 0.0f, s1 = 0.0f;
  for (int d = 0; d < DD; ++d) {
    float qv = q[d];
    s0 += qv * gw[(h * DD + d) * 3 + 0];
    s1 += qv * gw[(h * DD + d) * 3 + 1];
  }
  gcmp[t] = 1.0f / (1.0f + __expf(-s0));
  gslc[t] = 1.0f / (1.0f + __expf(-s1));
}

// ---- Kernel 3: compressed branch + top-16 block selection (1 wave / tile) ---
__global__ void __launch_bounds__(32)
cmp_sel_kernel(const float* __restrict__ pq, const float* __restrict__ kc,
               const float* __restrict__ vc, const float* __restrict__ gcmp,
               const int* __restrict__ xoff, unsigned* __restrict__ selmask,
               float* __restrict__ od) {
  __shared__ __align__(16) float Qs[QT][DD];       // 4 KB
  __shared__ __align__(16) float Kcs[NB][DD];      // 8 KB
  __shared__ __align__(16) float Vcs[NB][DD];      // 8 KB
  __shared__ __align__(16) float Sc[QT][NB];       // causal-masked scores
  __shared__ __align__(16) float Pc[QT][NB];       // silu probs
  __shared__ unsigned Um[2][QT];

  int bid  = blockIdx.x;
  int qt   = bid & (NQT - 1);
  int h    = (bid >> 6) & (HH - 1);
  int b    = bid >> 9;
  int lane = threadIdx.x;
  int half = lane >> 4;              // 0..1
  int lm   = lane & 15;

  int qblk = qt >> 1;                // uniform: 16-query tile sits in one block
  int len  = xoff[b + 1] - xoff[b];
  int clen = (len + BS - 1) / BS;
  const float scale = 0.125f;        // D^-0.5

  // async DMA the compressed K/V head slab into LDS (ASYNCcnt path)
  const float* kch = kc + (size_t)(b * HH + h) * NB * DD;
  const float* vch = vc + (size_t)(b * HH + h) * NB * DD;
  {
    unsigned lk = lds_off(&Kcs[0][0]);
    unsigned lv = lds_off(&Vcs[0][0]);
    for (int i = 0; i < (NB * DD) / (4 * 32); ++i) {   // 16 issues x 512B
      int idx = i * 32 + lane;                         // float4 index
      async_ld16(lk + idx * 16, kch + idx * 4);
      async_ld16(lv + idx * 16, vch + idx * 4);
    }
  }
  // Q tile via regular vector loads (overlaps with the async DMA above)
  for (int i = lane; i < QT * DD; i += 32) {
    int m = i >> 6, d = i & 63;
    Qs[m][d] = pq[((size_t)(b * NN + qt * QT + m) * HH + h) * DD + d];
  }
  wait_async0();
  __syncthreads();

  // S(16x32) = Q(16x64) @ Kc^T(64x32): two 16x16 tiles, K=64 in 16 steps
  v8f acc0 = {}, acc1 = {};
  for (int kk = 0; kk < 16; ++kk) {
    int k0 = 4 * kk + 2 * half;
    v2f a;  a.x  = Qs[lm][k0];        a.y  = Qs[lm][k0 + 1];
    v2f b0; b0.x = Kcs[lm][k0];       b0.y = Kcs[lm][k0 + 1];       // B[k][n]=Kc[n][k]
    v2f b1; b1.x = Kcs[16 + lm][k0];  b1.y = Kcs[16 + lm][k0 + 1];
    acc0 = wmma4(a, b0, acc0);
    acc1 = wmma4(a, b1, acc1);
  }

#pragma unroll
  for (int r = 0; r < 8; ++r) {
    int m = r + 8 * half;            // C/D layout: M = r + 8*half, N = lane&15
    {
      float s = acc0[r] * scale; int j = lm;
      bool causal = (j <= qblk);
      Sc[m][j] = causal ? s : NEG_INF;
      Pc[m][j] = (causal && j < clen) ? silu_f(s) : 0.0f;
    }
    {
      float s = acc1[r] * scale; int j = 16 + lm;
      bool causal = (j <= qblk);
      Sc[m][j] = causal ? s : NEG_INF;
      Pc[m][j] = (causal && j < clen) ? silu_f(s) : 0.0f;
    }
  }
  __syncthreads();

  // exact top-16 via rank (value desc, index asc ties) == jax.lax.top_k
  unsigned m16 = 0u;
  for (int jj = 0; jj < 16; ++jj) {
    int j = half * 16 + jj;
    if (j <= qblk) {
      float sj = Sc[lm][j];
      int rank = 0;
      for (int i = 0; i < NB; ++i) {
        float si = Sc[lm][i];
        rank += (si > sj) || (si == sj && i < j);
      }
      if (rank < TOPS) m16 |= (1u << jj);
    }
  }
  Um[half][lm] = m16;
  __syncthreads();
  if (lane < QT)
    selmask[(size_t)(b * HH + h) * NN + qt * QT + lane] =
        Um[0][lane] | (Um[1][lane] << 16);

  // O_cmp(16x64) = P(16x32) @ Vc(32x64): 4 N-tiles, K=32 in 8 steps
  v8f oc0 = {}, oc1 = {}, oc2 = {}, oc3 = {};
  for (int kk = 0; kk < 8; ++kk) {
    int k0 = 4 * kk + 2 * half;
    v2f a; a.x = Pc[lm][k0]; a.y = Pc[lm][k0 + 1];
    v2f bv;
    bv.x = Vcs[k0][lm];      bv.y = Vcs[k0 + 1][lm];      oc0 = wmma4(a, bv, oc0);
    bv.x = Vcs[k0][16 + lm]; bv.y = Vcs[k0 + 1][16 + lm]; oc1 = wmma4(a, bv, oc1);
    bv.x = Vcs[k0][32 + lm]; bv.y = Vcs[k0 + 1][32 + lm]; oc2 = wmma4(a, bv, oc2);
    bv.x = Vcs[k0][48 + lm]; bv.y = Vcs[k0 + 1][48 + lm]; oc3 = wmma4(a, bv, oc3);
  }
  v8f oc[4] = {oc0, oc1, oc2, oc3};
#pragma unroll
  for (int nt = 0; nt < 4; ++nt)
#pragma unroll
    for (int r = 0; r < 8; ++r) {
      int m = r + 8 * half;
      int q = qt * QT + m;
      int d = nt * 16 + lm;
      float g = gcmp[(size_t)(b * NN + q) * HH + h];
      od[((size_t)(b * NN + q) * HH + h) * DD + d] = oc[nt][r] * g;
    }
}

// ---- Kernel 4: selected branch (token causal over selected blocks) ----------
__global__ void __launch_bounds__(32)
slc_kernel(const float* __restrict__ pq, const float* __restrict__ pk,
           const float* __restrict__ pv, const float* __restrict__ gslc,
           const unsigned* __restrict__ selmask, float* __restrict__ od) {
  __shared__ __align__(16) float Qs[QT][DD];
  __shared__ __align__(16) float Ks[BS][DD];
  __shared__ __align__(16) float Vs[BS][DD];
  __shared__ __align__(16) float Ps[QT][BS];
  __shared__ unsigned Sm[QT];

  int bid  = blockIdx.x;
  int qt   = bid & (NQT - 1);
  int h    = (bid >> 6) & (HH - 1);
  int b    = bid >> 9;
  int lane = threadIdx.x;
  int half = lane >> 4;
  int lm   = lane & 15;
  int qblk = qt >> 1;
  const float scale = 0.125f;

  for (int i = lane; i < QT * DD; i += 32) {
    int m = i >> 6, d = i & 63;
    Qs[m][d] = pq[((size_t)(b * NN + qt * QT + m) * HH + h) * DD + d];
  }
  if (lane < QT)
    Sm[lane] = selmask[(size_t)(b * HH + h) * NN + qt * QT + lane];
  __syncthreads();

  unsigned uni = 0u;
  for (int m = 0; m < QT; ++m) uni |= Sm[m];   // uniform across the wave

  unsigned ldsK = lds_off(&Ks[0][0]);
  unsigned ldsV = lds_off(&Vs[0][0]);

  v8f o0 = {}, o1 = {}, o2 = {}, o3 = {};
  for (int j = 0; j <= qblk; ++j) {
    if (!((uni >> j) & 1u)) continue;          // uniform branch: EXEC stays full
    __syncthreads();
    const float* kb = pk + ((size_t)(b * NN + j * BS) * HH + h) * DD;
    const float* vb = pv + ((size_t)(b * NN + j * BS) * HH + h) * DD;
    // async DMA the 32x64 K and V tiles into LDS (rows strided by H*D in HBM)
    for (int i = 0; i < (BS * DD) / (4 * 32); ++i) {   // 16 issues x 512B each
      int idx = i * 32 + lane;                         // float4 index in tile
      int t = idx >> 4;                                // row (16 float4 / row)
      int c = idx & 15;                                // float4 within row
      async_ld16(ldsK + idx * 16, kb + (size_t)t * HH * DD + c * 4);
      async_ld16(ldsV + idx * 16, vb + (size_t)t * HH * DD + c * 4);
    }
    // prefetch next selected block while this one streams (global_prefetch_b8)
    {
      int j2 = j + 1;
      while (j2 <= qblk && !((uni >> j2) & 1u)) ++j2;
      if (j2 <= qblk) {
        const float* nk = pk + ((size_t)(b * NN + j2 * BS) * HH + h) * DD;
        const float* nv = pv + ((size_t)(b * NN + j2 * BS) * HH + h) * DD;
        __builtin_prefetch(nk + (size_t)lane * HH * DD, 0, 1);
        __builtin_prefetch(nv + (size_t)lane * HH * DD, 0, 1);
      }
    }
    wait_async0();
    __syncthreads();

    // S(16x32) = Q @ K_blk^T
    v8f s0 = {}, s1 = {};
    for (int kk = 0; kk < 16; ++kk) {
      int k0 = 4 * kk + 2 * half;
      v2f a;  a.x  = Qs[lm][k0];       a.y  = Qs[lm][k0 + 1];
      v2f b0; b0.x = Ks[lm][k0];       b0.y = Ks[lm][k0 + 1];
      v2f b1; b1.x = Ks[16 + lm][k0];  b1.y = Ks[16 + lm][k0 + 1];
      s0 = wmma4(a, b0, s0);
      s1 = wmma4(a, b1, s1);
    }
#pragma unroll
    for (int r = 0; r < 8; ++r) {
      int m = r + 8 * half;
      int q = qt * QT + m;
      bool sel = (Sm[m] >> j) & 1u;
      { int t = lm;      int kp = j * BS + t;
        Ps[m][t] = (sel && q >= kp) ? silu_f(s0[r] * scale) : 0.0f; }
      { int t = 16 + lm; int kp = j * BS + t;
        Ps[m][t] = (sel && q >= kp) ? silu_f(s1[r] * scale) : 0.0f; }
    }
    __syncthreads();

    // O += P(16x32) @ V_blk(32x64)
    for (int kk = 0; kk < 8; ++kk) {
      int k0 = 4 * kk + 2 * half;
      v2f a; a.x = Ps[lm][k0]; a.y = Ps[lm][k0 + 1];
      v2f bv;
      bv.x = Vs[k0][lm];      bv.y = Vs[k0 + 1][lm];      o0 = wmma4(a, bv, o0);
      bv.x = Vs[k0][16 + lm]; bv.y = Vs[k0 + 1][16 + lm]; o1 = wmma4(a, bv, o1);
      bv.x = Vs[k0][32 + lm]; bv.y = Vs[k0 + 1][32 + lm]; o2 = wmma4(a, bv, o2);
      bv.x = Vs[k0][48 + lm]; bv.y = Vs[k0 + 1][48 + lm]; o3 = wmma4(a, bv, o3);
    }
  }

  v8f oo[4] = {o0, o1, o2, o3};
#pragma unroll
  for (int nt = 0; nt < 4; ++nt)
#pragma unroll
    for (int r = 0; r < 8; ++r) {
      int m = r + 8 * half;
      int q = qt * QT + m;
      int d = nt * 16 + lm;
      float g = gslc[(size_t)(b * NN + q) * HH + h];
      size_t idx = ((size_t)(b * NN + q) * HH + h) * DD + d;
      od[idx] += oo[nt][r] * g;   // exclusive owner per element: no atomics
    }
}

// ---------------- Kernel 5: gather valid tokens -----------------------------
__global__ void gather_kernel(const float* __restrict__ od,
                              const int* __restrict__ gidx,
                              float* __restrict__ out, int tot) {
  int t = blockIdx.x * blockDim.x + threadIdx.x;
  if (t >= tot) return;
  int i = t / (HH * DD);
  int rem = t % (HH * DD);
  out[t] = od[(size_t)gidx[i] * HH * DD + rem];
}

extern "C" void kernel_launch(void* const* d_in, const int* in_sizes, int n_in,
                              void* d_out, int out_size, void* d_ws, size_t ws_size,
                              hipStream_t stream) {
  (void)n_in; (void)out_size; (void)ws_size;
  const float* pq   = (const float*)d_in[4];   // padded_q
  const float* pk   = (const float*)d_in[5];   // padded_k
  const float* pv   = (const float*)d_in[6];   // padded_v
  const int*   xoff = (const int*)d_in[7];     // x_offsets
  const float* gw   = (const float*)d_in[8];   // gate_w
  const int*   gidx = (const int*)d_in[10];    // gather_idx
  int ntok = in_sizes[10];

  float* ws   = (float*)d_ws;
  float* kc   = ws;                              // B*H*nb*D
  float* vc   = kc + BB * HH * NB * DD;
  float* gcmp = vc + BB * HH * NB * DD;          // B*N*H
  float* gslc = gcmp + BB * NN * HH;
  unsigned* selmask = (unsigned*)(gslc + BB * NN * HH);  // B*H*N
  float* od   = (float*)(selmask + BB * HH * NN);        // B*N*H*D

  kcmp_vcmp_kernel<<<(BB * HH * NB * DD) / 256, 256, 0, stream>>>(pk, pv, kc, vc);
  gates_kernel<<<(BB * NN * HH) / 256, 256, 0, stream>>>(pq, gw, gcmp, gslc);
  cmp_sel_kernel<<<BB * HH * NQT, 32, 0, stream>>>(pq, kc, vc, gcmp, xoff, selmask, od);
  slc_kernel<<<BB * HH * NQT, 32, 0, stream>>>(pq, pk, pv, gslc, selmask, od);
  int tot = ntok * HH * DD;
  gather_kernel<<<(tot + 255) / 256, 256, 0, stream>>>(od, gidx, (float*)d_out, tot);
}